// AttentionMechanism_65859028517400
// MI455X (gfx1250) — compile-verified
//
#include <hip/hip_runtime.h>
#include <hip/hip_bf16.h>

typedef __attribute__((ext_vector_type(2))) float v2f;
typedef __attribute__((ext_vector_type(4))) float v4f;
typedef __attribute__((ext_vector_type(8))) float v8f;

// Problem constants (from the reference)
#define DD 512
#define KK 10
#define LL 256
#define VV 80

__global__ __launch_bounds__(512) void graves_attn_kernel(
    const float* __restrict__ inputs,      // [B, D]
    const float* __restrict__ prev_kappa,  // [B, K]
    const float* __restrict__ OH,          // [B, L, V] one-hot
    const int*   __restrict__ seqlen,      // [B]
    const float* __restrict__ W,           // [D, 3K]
    const float* __restrict__ bias,        // [3K]
    const float* __restrict__ kscale_p,    // [1]
    float* __restrict__ out_w,             // [B, V]
    float* __restrict__ out_kappa)         // [B, K]
{
    __shared__ float s_alpha[16][KK];
    __shared__ float s_beta [16][KK];
    __shared__ float s_kappa[16][KK];
    __shared__ float s_phi  [16][LL];

    const int tid    = threadIdx.x;
    const int lane   = tid & 31;
    const int wave   = tid >> 5;
    const int m_base = blockIdx.x * 16;

    // ---------- Phase A: raw = inputs @ W + b via fp32 WMMA (wave 0 only) ----------
    if (wave == 0) {
        const int r    = lane & 15;         // A: row index / B: column index
        const int koff = (lane >> 4) * 2;   // k sub-offset within the 4-wide K slab
        const int n1   = 16 + r;            // second N tile column (valid if < 30)
        // Branchless guard: clamp address, zero the contribution via multiply.
        const int   n1c = (n1 < 3 * KK) ? n1 : (3 * KK - 1);
        const float m1  = (n1 < 3 * KK) ? 1.0f : 0.0f;
        const float* arow = inputs + (size_t)(m_base + r) * DD;

        v8f c0 = {}; v8f c1 = {};
        for (int k = 0; k < DD; k += 4) {
            // A-matrix 16x4 f32: VGPR0 = K(koff), VGPR1 = K(koff+1)
            v2f a;
            a.x = arow[k + koff];
            a.y = arow[k + koff + 1];
            // B-matrix 4x16 f32: lane holds column r / 16+r, rows k+koff, k+koff+1
            const float* wr0 = W + (size_t)(k + koff    ) * (3 * KK);
            const float* wr1 = W + (size_t)(k + koff + 1) * (3 * KK);
            v2f b0, b1;
            b0.x = wr0[r];
            b0.y = wr1[r];
            b1.x = wr0[n1c] * m1;           // unconditional load + v_mul (no EXEC branches)
            b1.y = wr1[n1c] * m1;
            c0 = __builtin_amdgcn_wmma_f32_16x16x4_f32(false, a, false, b0,
                                                       (short)0, c0, false, false);
            c1 = __builtin_amdgcn_wmma_f32_16x16x4_f32(false, a, false, b1,
                                                       (short)0, c1, false, false);
        }

        const float kscale = kscale_p[0];
        const float bias0  = bias[r];
        const float bias1  = bias[n1c] * m1;

        // C/D layout: VGPR i -> row = i + (lane/16)*8, col = lane%16 (+16 for tile 1)
        #pragma unroll
        for (int i = 0; i < 8; ++i) {
            const int row = i + (lane >> 4) * 8;
            const int gb  = m_base + row;
            // Tile 0: cols 0..15 -> alpha_hat (0..9), beta_hat (10..15)
            {
                const float raw = c0[i] + bias0;
                const float e   = __expf(fminf(fmaxf(raw, -8.0f), 8.0f));
                if (r < KK) s_alpha[row][r] = e;
                else        s_beta [row][r - KK] = e;
            }
            // Tile 1: cols 16..29 -> beta_hat (16..19), kappa_hat (20..29)
            if (n1 < 3 * KK) {
                const float raw = c1[i] + bias1;
                if (n1 < 2 * KK) {
                    s_beta[row][n1 - KK] = __expf(fminf(fmaxf(raw, -8.0f), 8.0f));
                } else {
                    const int kk = n1 - 2 * KK;
                    const float dk  = __expf(fminf(fmaxf(raw + kscale, -8.0f), 5.0f));
                    const float kap = prev_kappa[(size_t)gb * KK + kk] + dk;
                    s_kappa[row][kk] = kap;
                    out_kappa[(size_t)gb * KK + kk] = kap;
                }
            }
        }
    }
    __syncthreads();

    // ---------- Phase B: wave w owns batch row (m_base + w) ----------
    const int row = wave;
    const int gb  = m_base + row;

    float al[KK], be[KK], ka[KK];
    #pragma unroll
    for (int k = 0; k < KK; ++k) {
        al[k] = s_alpha[row][k];
        be[k] = s_beta [row][k];
        ka[k] = s_kappa[row][k];
    }
    const int sl = seqlen[gb];

    // Stage 1: phi[l] for l = lane + 32*j, masked by sequence length
    #pragma unroll
    for (int j = 0; j < LL / 32; ++j) {
        const int   l = lane + j * 32;
        const float u = (float)(l + 1);
        float acc = 0.0f;
        #pragma unroll
        for (int k = 0; k < KK; ++k) {
            const float t  = ka[k] - u;
            const float ex = fminf(fmaxf(-be[k] * t * t, -50.0f), 0.0f);
            acc += al[k] * __expf(ex);
        }
        s_phi[row][l] = (l < sl) ? acc : 0.0f;
    }
    __syncthreads();

    // Stage 2: streaming einsum w[b,v] = sum_l phi[l] * OH[b,l,v]
    // Lanes 0..19 each own 4 consecutive v's (b128 coalesced, non-temporal).
    const int  vbase  = lane * 4;
    const bool active = (vbase < VV);
    const float* ohrow = OH + (size_t)gb * LL * VV;

    v4f acc = {};
    if (active) {
        for (int l = 0; l < LL; ++l) {
            const float phi = s_phi[row][l];   // LDS broadcast read
            v4f oh = __builtin_nontemporal_load(
                (const v4f*)(ohrow + (size_t)l * VV + vbase));
            acc += oh * phi;
        }
        *(v4f*)(out_w + (size_t)gb * VV + vbase) = acc;
    }
}

extern "C" void kernel_launch(void* const* d_in, const int* in_sizes, int n_in,
                              void* d_out, int out_size, void* d_ws, size_t ws_size,
                              hipStream_t stream) {
    const float* inputs     = (const float*)d_in[0];
    const float* prev_kappa = (const float*)d_in[1];
    const float* OH         = (const float*)d_in[2];
    const int*   seqlen     = (const int*)  d_in[3];
    const float* W          = (const float*)d_in[4];
    const float* bias       = (const float*)d_in[5];
    const float* kscale     = (const float*)d_in[6];

    const int B = in_sizes[3];            // sequence_lengths has one entry per batch row
    float* out_w     = (float*)d_out;     // [B, V] first
    float* out_kappa = out_w + (size_t)B * VV;  // then [B, K]

    dim3 grid(B / 16), block(512);
    graves_attn_kernel<<<grid, block, 0, stream>>>(
        inputs, prev_kappa, OH, seqlen, W, bias, kscale, out_w, out_kappa);
}